// GUNet_17944373363041
// MI455X (gfx1250) — compile-verified
//
#include <hip/hip_runtime.h>
#include <cstddef>

// ---------------------------------------------------------------------------
// Graph U-Net forward on MI455X (gfx1250).
//  - All dense GEMMs on the WMMA f32 pipe (v_wmma_f32_16x16x4_f32), 128x64
//    macro tiles, 8 waves x (2x2) 16x16 accumulators.
//  - Global->LDS staging via CDNA5 async copies (global_load_async_to_lds_b128,
//    ASYNCcnt double-buffer pipeline, branch-free steady state).
//  - ~190 GFLOP fp32 vs ~0.5 GB unique traffic @ 23.3 TB/s -> compute bound,
//    adjacency (64 MB) is L2-resident (192 MB).
// ---------------------------------------------------------------------------

typedef __attribute__((ext_vector_type(2))) float v2f;
typedef __attribute__((ext_vector_type(8))) float v8f;

#define TM 128
#define TN 64
#define TK 16
#define SA 20  // A-tile LDS row stride (floats): even (b64 frags), %4==0 (b128 async)

// issue one 16B async global->LDS copy (tracked by ASYNCcnt)
__device__ __forceinline__ void async_b128(const float* lds, const float* g) {
  asm volatile("global_load_async_to_lds_b128 %0, %1, off"
               :: "v"((unsigned)(size_t)lds),
                  "v"((unsigned long long)(size_t)g)
               : "memory");
}
__device__ __forceinline__ void wait_async_le3() {
  asm volatile("s_wait_asynccnt 3" ::: "memory");
}

// C = op(A) @ op(B), row-major; DIAG!=0 computes (A+I)@(B+I) (augment_adj).
// Requires M%128==0, N%64==0, K%16==0, lda/ldb multiples of 4.
template <int DIAG>
__global__ __launch_bounds__(256) void gemm_f32_wmma(
    const float* __restrict__ A, const float* __restrict__ B,
    float* __restrict__ C, int M, int N, int K, int lda, int ldb, int ldc)
{
  __shared__ __align__(16) float As[2][TM * SA];   // [m][k], stride SA
  __shared__ __align__(16) float Bs[2][TK * TN];   // [k][n]

  const int tid  = threadIdx.x;
  const int lane = tid & 31;
  const int wid  = tid >> 5;
  const int wm   = (wid & 3) * 32;   // wave M offset (2 subtiles: wm, wm+16)
  const int wn   = (wid >> 2) * 32;  // wave N offset (2 subtiles: wn, wn+16)
  const int half = lane >> 4;
  const int lm   = lane & 15;

  const int row0 = blockIdx.y * TM;
  const int col0 = blockIdx.x * TN;

  // async tile-load slots: A 128x16 = 2 x b128/thread, B 16x64 = 1 x b128/thread
  const int am = tid >> 1;          // 0..127
  const int ak = (tid & 1) * 8;     // 0 or 8
  const int bk = tid >> 4;          // 0..15
  const int bn = (tid & 15) * 4;    // 0..60

  auto issue = [&](int buf, int k0) {
    const float* ga = A + (size_t)(row0 + am) * lda + (k0 + ak);
    async_b128(&As[buf][am * SA + ak],     ga);
    async_b128(&As[buf][am * SA + ak + 4], ga + 4);
    async_b128(&Bs[buf][bk * TN + bn],
               B + (size_t)(k0 + bk) * ldb + (col0 + bn));
  };

  v8f acc00 = {}, acc01 = {}, acc10 = {}, acc11 = {};

  issue(0, 0);
  int buf = 0;
  for (int k0 = 0; k0 < K; k0 += TK) {
    // Branch-free pipeline: on the final slab, re-issue the same (in-bounds)
    // slab into the dead buffer so ASYNCcnt<=3 always means "previous tile
    // landed".  The dangling ops drain at S_ENDPGM's implicit wait-idle.
    const int kn = (k0 + TK < K) ? (k0 + TK) : k0;
    issue(buf ^ 1, kn);
    wait_async_le3();
    __syncthreads();

    if (DIAG) {  // (A+I)/(B+I): touch the <=16 diagonal entries of this K-slab
      if (tid < TK) {
        const int kg = k0 + tid;
        const int ma = kg - row0;
        if (ma >= 0 && ma < TM) As[buf][ma * SA + tid] += 1.0f;
        const int nb = kg - col0;
        if (nb >= 0 && nb < TN) Bs[buf][tid * TN + nb] += 1.0f;
      }
      __syncthreads();
    }

#pragma unroll
    for (int kk = 0; kk < TK; kk += 4) {
      // fp32 fragment layouts (ISA 7.12.2): lanes0-15 K=kk,kk+1; lanes16-31
      // K=kk+2,kk+3; M/N = lane%16.  A frag = one aligned b64 LDS read.
      const int krow = kk + 2 * half;
      v2f a0 = *(const v2f*)&As[buf][(wm + lm) * SA + krow];
      v2f a1 = *(const v2f*)&As[buf][(wm + 16 + lm) * SA + krow];
      v2f b0, b1;
      b0.x = Bs[buf][(krow + 0) * TN + wn + lm];
      b0.y = Bs[buf][(krow + 1) * TN + wn + lm];
      b1.x = Bs[buf][(krow + 0) * TN + wn + 16 + lm];
      b1.y = Bs[buf][(krow + 1) * TN + wn + 16 + lm];
      acc00 = __builtin_amdgcn_wmma_f32_16x16x4_f32(false, a0, false, b0,
                                                    (short)0, acc00, false, false);
      acc01 = __builtin_amdgcn_wmma_f32_16x16x4_f32(false, a0, false, b1,
                                                    (short)0, acc01, false, false);
      acc10 = __builtin_amdgcn_wmma_f32_16x16x4_f32(false, a1, false, b0,
                                                    (short)0, acc10, false, false);
      acc11 = __builtin_amdgcn_wmma_f32_16x16x4_f32(false, a1, false, b1,
                                                    (short)0, acc11, false, false);
    }
    __syncthreads();
    buf ^= 1;
  }

  // C/D layout: VGPR v -> M = v + 8*half, N = lane%16
#pragma unroll
  for (int v = 0; v < 8; ++v) {
    const size_t r0 = (size_t)(row0 + wm + v + 8 * half);
    const size_t r1 = r0 + 16;
    C[r0 * ldc + col0 + wn + lm]      = acc00[v];
    C[r0 * ldc + col0 + wn + 16 + lm] = acc01[v];
    C[r1 * ldc + col0 + wn + lm]      = acc10[v];
    C[r1 * ldc + col0 + wn + 16 + lm] = acc11[v];
  }
}

// ---------------------------------------------------------------------------
// elementwise / reduction helpers
// ---------------------------------------------------------------------------
__global__ void k_zero(float* p, long n) {
  long i = (long)blockIdx.x * blockDim.x + threadIdx.x;
  if (i < n) p[i] = 0.0f;
}

__global__ void k_scatter_edges(float* A, const int* __restrict__ ei, int E, int n) {
  int e = blockIdx.x * blockDim.x + threadIdx.x;
  if (e < E) {
    int s = ei[e], d = ei[E + e];
    A[(size_t)s * n + d] = 1.0f;  // .set semantics: duplicates all write 1.0
  }
}

// dinv[row] = rsqrt(rowsum(A) + 2)   (Ahat = A + 2I)
__global__ void k_rowsum_dinv(const float* __restrict__ A, int n, float* dinv) {
  __shared__ float red[256];
  const int row = blockIdx.x;
  float s = 0.0f;
  for (int j = threadIdx.x; j < n; j += 256) s += A[(size_t)row * n + j];
  red[threadIdx.x] = s;
  __syncthreads();
  for (int o = 128; o > 0; o >>= 1) {
    if (threadIdx.x < o) red[threadIdx.x] += red[threadIdx.x + o];
    __syncthreads();
  }
  if (threadIdx.x == 0) {
    float d = red[0] + 2.0f;
    dinv[row] = d > 0.0f ? rsqrtf(d) : 0.0f;
  }
}

__global__ void k_rowscale(float* out, const float* __restrict__ in,
                           const float* __restrict__ dinv, int rows, int cols) {
  long i = (long)blockIdx.x * blockDim.x + threadIdx.x;
  if (i < (long)rows * cols) out[i] = in[i] * dinv[i / cols];
}

// out = dinv[r] * (Y + 2Z) + b[c], optional relu
__global__ void k_gcn_epilogue(float* out, const float* __restrict__ Y,
                               const float* __restrict__ Z,
                               const float* __restrict__ dinv,
                               const float* __restrict__ b,
                               int rows, int cols, int relu) {
  long i = (long)blockIdx.x * blockDim.x + threadIdx.x;
  if (i < (long)rows * cols) {
    int r = (int)(i / cols), c = (int)(i % cols);
    float v = dinv[r] * (Y[i] + 2.0f * Z[i]) + b[c];
    out[i] = relu ? fmaxf(v, 0.0f) : v;
  }
}

__global__ void k_zero_diag(float* A, int n) {
  int i = blockIdx.x * blockDim.x + threadIdx.x;
  if (i < n) A[(size_t)i * n + i] = 0.0f;
}

__global__ void k_inv_norm(const float* __restrict__ w, int len, float* out) {
  __shared__ float red[256];
  float s = 0.0f;
  for (int i = threadIdx.x; i < len; i += 256) s += w[i] * w[i];
  red[threadIdx.x] = s;
  __syncthreads();
  for (int o = 128; o > 0; o >>= 1) {
    if (threadIdx.x < o) red[threadIdx.x] += red[threadIdx.x + o];
    __syncthreads();
  }
  if (threadIdx.x == 0) out[0] = rsqrtf(red[0]);
}

// score[row] = tanh(dot(x[row], w) * inv_norm); one wave32 per row
__global__ void k_score(const float* __restrict__ x, const float* __restrict__ w,
                        const float* __restrict__ invn, int n, int cols, float* sc) {
  const int row  = blockIdx.x * 8 + (threadIdx.x >> 5);
  const int lane = threadIdx.x & 31;
  if (row < n) {
    float s = 0.0f;
    for (int c = lane; c < cols; c += 32) s += x[(size_t)row * cols + c] * w[c];
    for (int o = 16; o > 0; o >>= 1) s += __shfl_xor(s, o, 32);
    if (lane == 0) sc[row] = tanhf(s * invn[0]);
  }
}

// single-workgroup bitonic sort (descending) over n<=4096 scores; emit top-k idx.
__global__ __launch_bounds__(1024) void k_bitonic_topk(
    const float* __restrict__ sc, int n, int k, int* perm) {
  __shared__ float key[4096];
  __shared__ int   idx[4096];
  for (int i = threadIdx.x; i < n; i += blockDim.x) { key[i] = sc[i]; idx[i] = i; }
  __syncthreads();
  for (int len = 2; len <= n; len <<= 1) {
    for (int j = len >> 1; j > 0; j >>= 1) {
      for (int i = threadIdx.x; i < n; i += blockDim.x) {
        int ixj = i ^ j;
        if (ixj > i) {
          bool desc = ((i & len) == 0);
          bool sw = desc ? (key[i] < key[ixj]) : (key[i] > key[ixj]);
          if (sw) {
            float tk = key[i]; key[i] = key[ixj]; key[ixj] = tk;
            int   ti = idx[i]; idx[i] = idx[ixj]; idx[ixj] = ti;
          }
        }
      }
      __syncthreads();
    }
  }
  for (int i = threadIdx.x; i < k; i += blockDim.x) perm[i] = idx[i];
}

__global__ void k_gather_x(float* xo, const float* __restrict__ xi,
                           const int* __restrict__ perm,
                           const float* __restrict__ sc, int k, int cols) {
  long i = (long)blockIdx.x * blockDim.x + threadIdx.x;
  if (i < (long)k * cols) {
    int r = (int)(i / cols), c = (int)(i % cols);
    int p = perm[r];
    xo[i] = xi[(size_t)p * cols + c] * sc[p];
  }
}

__global__ void k_gather_A(float* Ao, const float* __restrict__ Ai,
                           const int* __restrict__ perm, int k, int n) {
  int j = blockIdx.x * blockDim.x + threadIdx.x;
  int i = blockIdx.y;
  if (j < k) Ao[(size_t)i * k + j] = Ai[(size_t)perm[i] * n + perm[j]];
}

__global__ void k_copy(float* o, const float* __restrict__ s, long n) {
  long i = (long)blockIdx.x * blockDim.x + threadIdx.x;
  if (i < n) o[i] = s[i];
}

// out[perm[r], c] += xc[r, c]   (perm entries unique -> race-free)
__global__ void k_scatter_add(float* out, const float* __restrict__ xc,
                              const int* __restrict__ perm, int k, int cols) {
  long i = (long)blockIdx.x * blockDim.x + threadIdx.x;
  if (i < (long)k * cols) {
    int r = (int)(i / cols), c = (int)(i % cols);
    out[(size_t)perm[r] * cols + c] += xc[i];
  }
}

// ---------------------------------------------------------------------------
static inline int div_up_l(long a, long b) { return (int)((a + b - 1) / b); }

extern "C" void kernel_launch(void* const* d_in, const int* in_sizes, int n_in,
                              void* d_out, int out_size, void* d_ws, size_t ws_size,
                              hipStream_t stream) {
  (void)n_in; (void)out_size; (void)ws_size;
  const int N0 = 4096, HIDC = 256, OUTC = 64, INC = 512;

  const float* x   = (const float*)d_in[0];
  const int*   ei  = (const int*)d_in[1];
  const float* dW[4] = {(const float*)d_in[2], (const float*)d_in[3],
                        (const float*)d_in[4], (const float*)d_in[5]};
  const float* db[4] = {(const float*)d_in[6], (const float*)d_in[7],
                        (const float*)d_in[8], (const float*)d_in[9]};
  const float* pw[3] = {(const float*)d_in[10], (const float*)d_in[11],
                        (const float*)d_in[12]};
  const float* uW[3] = {(const float*)d_in[13], (const float*)d_in[14],
                        (const float*)d_in[15]};
  const float* ub[3] = {(const float*)d_in[16], (const float*)d_in[17],
                        (const float*)d_in[18]};
  const int E = in_sizes[1] / 2;

  // ---- bump allocator over workspace ----
  float* wsf = (float*)d_ws;
  size_t off = 0;
  auto alloc = [&](size_t nf) -> float* {
    float* p = wsf + off;
    off += (nf + 63) & ~(size_t)63;
    return p;
  };
  float* A0  = alloc((size_t)N0 * N0);
  float* AUG = alloc((size_t)N0 * N0);     // augment_adj scratch (reused per level)
  float* A1  = alloc((size_t)2048 * 2048);
  float* A2  = alloc((size_t)1024 * 1024);
  float* A3  = alloc((size_t)512 * 512);
  float* xs0 = alloc((size_t)N0 * HIDC);
  float* xs1 = alloc((size_t)2048 * HIDC);
  float* xs2 = alloc((size_t)1024 * HIDC);
  float* T0  = alloc((size_t)N0 * HIDC);   // X@W then Z
  float* T1  = alloc((size_t)N0 * HIDC);   // A@Z
  float* XP  = alloc((size_t)2048 * HIDC); // pooled features (gcn input)
  float* XU  = alloc((size_t)N0 * HIDC);   // unpool combine buffer
  float* XA  = alloc((size_t)N0 * HIDC);   // current features on up path
  float* dv0 = alloc(N0);
  float* dv1 = alloc(2048);
  float* dv2 = alloc(1024);
  float* dv3 = alloc(512);
  float* sco = alloc(N0);
  float* inv = alloc(64);
  int* perm1 = (int*)alloc(2048);
  int* perm2 = (int*)alloc(1024);
  int* perm3 = (int*)alloc(512);

  auto gemm = [&](const float* Am, const float* Bm, float* Cm,
                  int M, int Nn, int K, int lda, int ldb, int ldc, int diag) {
    dim3 g(Nn / TN, M / TM);
    if (diag)
      gemm_f32_wmma<1><<<g, 256, 0, stream>>>(Am, Bm, Cm, M, Nn, K, lda, ldb, ldc);
    else
      gemm_f32_wmma<0><<<g, 256, 0, stream>>>(Am, Bm, Cm, M, Nn, K, lda, ldb, ldc);
  };

  // y = dinv * (A @ (dinv*(X@W)) + 2*dinv*(X@W)) + b
  auto gcn = [&](const float* X, int n, int fin, int fout, const float* Am,
                 const float* dinv, const float* W, const float* b, float* out,
                 int relu) {
    gemm(X, W, T0, n, fout, fin, fin, fout, fout, 0);              // T0 = X@W
    long cnt = (long)n * fout;
    k_rowscale<<<div_up_l(cnt, 256), 256, 0, stream>>>(T0, T0, dinv, n, fout); // Z
    gemm(Am, T0, T1, n, fout, n, n, fout, fout, 0);                // T1 = A@Z
    k_gcn_epilogue<<<div_up_l(cnt, 256), 256, 0, stream>>>(out, T1, T0, dinv, b,
                                                           n, fout, relu);
  };

  // augment_adj + topk pool: Aout(kxk), perm(k), xout(k x HID)
  auto pool = [&](const float* Ain, int n, const float* xin, const float* w,
                  float* Aout, int* perm, float* xout) {
    int k = n / 2;
    gemm(Ain, Ain, AUG, n, n, n, n, n, n, 1);                      // (A+I)@(A+I)
    k_zero_diag<<<div_up_l(n, 256), 256, 0, stream>>>(AUG, n);
    k_inv_norm<<<1, 256, 0, stream>>>(w, HIDC, inv);
    k_score<<<n / 8, 256, 0, stream>>>(xin, w, inv, n, HIDC, sco);
    k_bitonic_topk<<<1, 1024, 0, stream>>>(sco, n, k, perm);
    k_gather_x<<<div_up_l((long)k * HIDC, 256), 256, 0, stream>>>(xout, xin, perm,
                                                                  sco, k, HIDC);
    dim3 ga(div_up_l(k, 256), k);
    k_gather_A<<<ga, 256, 0, stream>>>(Aout, AUG, perm, k, n);
  };

  // ================= forward =================
  k_zero<<<div_up_l((long)N0 * N0, 256), 256, 0, stream>>>(A0, (long)N0 * N0);
  k_scatter_edges<<<div_up_l(E, 256), 256, 0, stream>>>(A0, ei, E, N0);
  k_rowsum_dinv<<<N0, 256, 0, stream>>>(A0, N0, dv0);
  gcn(x, N0, INC, HIDC, A0, dv0, dW[0], db[0], xs0, 1);

  pool(A0, 4096, xs0, pw[0], A1, perm1, XP);
  k_rowsum_dinv<<<2048, 256, 0, stream>>>(A1, 2048, dv1);
  gcn(XP, 2048, HIDC, HIDC, A1, dv1, dW[1], db[1], xs1, 1);

  pool(A1, 2048, xs1, pw[1], A2, perm2, XP);
  k_rowsum_dinv<<<1024, 256, 0, stream>>>(A2, 1024, dv2);
  gcn(XP, 1024, HIDC, HIDC, A2, dv2, dW[2], db[2], xs2, 1);

  pool(A2, 1024, xs2, pw[2], A3, perm3, XP);
  k_rowsum_dinv<<<512, 256, 0, stream>>>(A3, 512, dv3);
  gcn(XP, 512, HIDC, HIDC, A3, dv3, dW[3], db[3], XA, 1);          // XA: 512xHID

  // up i=0 (level 2): XU = xs2; XU[perm3] += XA; gcn -> XA (1024xHID), relu
  k_copy<<<div_up_l((long)1024 * HIDC, 256), 256, 0, stream>>>(XU, xs2,
                                                               (long)1024 * HIDC);
  k_scatter_add<<<div_up_l((long)512 * HIDC, 256), 256, 0, stream>>>(XU, XA, perm3,
                                                                     512, HIDC);
  gcn(XU, 1024, HIDC, HIDC, A2, dv2, uW[0], ub[0], XA, 1);

  // up i=1 (level 1)
  k_copy<<<div_up_l((long)2048 * HIDC, 256), 256, 0, stream>>>(XU, xs1,
                                                               (long)2048 * HIDC);
  k_scatter_add<<<div_up_l((long)1024 * HIDC, 256), 256, 0, stream>>>(XU, XA, perm2,
                                                                      1024, HIDC);
  gcn(XU, 2048, HIDC, HIDC, A1, dv1, uW[1], ub[1], XA, 1);

  // up i=2 (level 0): final, HID->OUT, no relu, straight into d_out
  k_copy<<<div_up_l((long)N0 * HIDC, 256), 256, 0, stream>>>(XU, xs0,
                                                             (long)N0 * HIDC);
  k_scatter_add<<<div_up_l((long)2048 * HIDC, 256), 256, 0, stream>>>(XU, XA, perm1,
                                                                      2048, HIDC);
  gcn(XU, N0, HIDC, OUTC, A0, dv0, uW[2], ub[2], (float*)d_out, 0);
}